// BoxAnchorAssigner_64441689309938
// MI455X (gfx1250) — compile-verified
//
#include <hip/hip_runtime.h>
#include <math.h>

// ---------------------------------------------------------------------------
// BoxAnchorAssigner for MI455X (gfx1250, wave32).
// GT table (128 x 12 floats: standup x1,y1,x2,y2, area, raw x,y,z,l,w,h,r)
// is staged into LDS with CDNA5 async global->LDS loads (ASYNCcnt-tracked).
// assign_main_kernel is placed first in the TU so the device-asm snippet
// shows the global_load_async_to_lds_b128 staging path.
// ---------------------------------------------------------------------------

#define MAX_GT 128

__constant__ const float kPOS_THR = 0.6f;
__constant__ const float kNEG_THR = 0.45f;

#define TWO_PI_F  6.28318530717958647692f
#define HALF_PI_F 1.57079632679489661923f

// Vector type matching the async-LDS builtin signature:
// __builtin_amdgcn_global_load_async_to_lds_b128(v4i AS1* src, v4i AS3* dst, Ii, Ii)
typedef int v4i_t __attribute__((vector_size(4 * sizeof(int))));
typedef v4i_t __attribute__((address_space(1)))* gptr_v4i;
typedef v4i_t __attribute__((address_space(3)))* lptr_v4i;

// gt slot layout (12 floats): [x1, y1, x2, y2, area, x, y, z, l, w, h, r]

__device__ __forceinline__ void encode_one(const float* __restrict__ anc7,
                                           const float* __restrict__ gslot,
                                           float reg[8], float dir[4]) {
    float xa = anc7[0], ya = anc7[1], za = anc7[2];
    float la = anc7[3], wa = anc7[4], ha = anc7[5], ra = anc7[6];
    float xg = gslot[5], yg = gslot[6], zg = gslot[7];
    float lg = gslot[8], wg = gslot[9], hg = gslot[10], rg = gslot[11];
    float diag = sqrtf(la * la + wa * wa);
    reg[0] = (xg - xa) / diag;
    reg[1] = (yg - ya) / diag;
    reg[2] = (zg - za) / ha;
    reg[3] = logf(lg / la);
    reg[4] = logf(wg / wa);
    reg[5] = logf(hg / ha);
    reg[6] = cosf(rg) - cosf(ra);
    reg[7] = sinf(rg) - sinf(ra);
    float rm = fmodf(rg, TWO_PI_F);
    if (rm < 0.0f) rm += TWO_PI_F;
    int q = (int)floorf(rm / HALF_PI_F);
    q = q < 0 ? 0 : (q > 3 ? 3 : q);
    dir[0] = dir[1] = dir[2] = dir[3] = 0.0f;
    dir[q] = 1.0f;
}

__device__ __forceinline__ float iou_su(float ax1, float ay1, float ax2, float ay2,
                                        float a1, const float* __restrict__ gslot) {
    float xext = fminf(ax2, gslot[2]) - fmaxf(ax1, gslot[0]) + 1.0f;
    float yext = fminf(ay2, gslot[3]) - fmaxf(ay1, gslot[1]) + 1.0f;
    float iou = 0.0f;
    if (xext > 0.0f && yext > 0.0f) {
        float inter = xext * yext;
        iou = inter / (a1 + gslot[4] - inter);
    }
    return iou;
}

// --------------------------- kernel: main assign ---------------------------
__global__ void __launch_bounds__(256)
assign_main_kernel(const float* __restrict__ anchors,
                   const float4* __restrict__ standup,
                   const float* __restrict__ wsgt,
                   int G, int A,
                   unsigned long long* __restrict__ winner,
                   float* __restrict__ out) {
    __shared__ __align__(16) float sgt[MAX_GT * 12];
    const int tid = threadIdx.x;
    const int nfl = G * 12;

    // ---- stage GT table into LDS with CDNA5 async global->LDS DMA ----
#if defined(__AMDGCN__) && __has_builtin(__builtin_amdgcn_global_load_async_to_lds_b128)
    {
        const int nchunk = (nfl * 4) >> 4;  // 16B chunks (12 floats/gt -> 48B, /16 exact)
        gptr_v4i gsrc = (gptr_v4i)(float*)wsgt;       // C-style cast drops const, AS1
        lptr_v4i ldst = (lptr_v4i)&sgt[0];            // generic->AS3
        for (int k = tid; k < nchunk; k += (int)blockDim.x) {
            __builtin_amdgcn_global_load_async_to_lds_b128(gsrc + k, ldst + k, 0, 0);
        }
#if __has_builtin(__builtin_amdgcn_s_wait_asynccnt)
        __builtin_amdgcn_s_wait_asynccnt(0);
#else
        asm volatile("s_wait_asynccnt 0x0" ::: "memory");
#endif
    }
#else
    for (int k = tid; k < nfl; k += (int)blockDim.x) sgt[k] = wsgt[k];
#endif
    __syncthreads();

    const int a = blockIdx.x * blockDim.x + tid;
    if (a >= A) return;

    const float4 sb = standup[a];
    const float ax1 = sb.x, ay1 = sb.y, ax2 = sb.z, ay2 = sb.w;
    const float a1 = (ax2 - ax1 + 1.0f) * (ay2 - ay1 + 1.0f);

    float best = -1.0f;
    int bg = 0;
    for (int g = 0; g < G; ++g) {
        const float* p = &sgt[g * 12];
        float iou = iou_su(ax1, ay1, ax2, ay2, a1, p);
        if (iou > best) { best = iou; bg = g; }   // first-max tie-break (ascending g)
        if (iou > 0.0f) {
            // key: max iou wins; equal iou -> smallest anchor index wins
            unsigned long long key =
                ((unsigned long long)__float_as_uint(iou) << 32) |
                (unsigned long long)(0xFFFFFFFFu - (unsigned)a);
            atomicMax(&winner[g], key);
        }
    }

    const bool pos = best > kPOS_THR;
    out[a] = pos ? 1.0f : (best < kNEG_THR ? 0.0f : -1.0f);

    float reg[8], dir[4];
    if (pos) {
        encode_one(anchors + (size_t)a * 7, &sgt[bg * 12], reg, dir);
    } else {
#pragma unroll
        for (int j = 0; j < 8; ++j) reg[j] = 0.0f;
#pragma unroll
        for (int j = 0; j < 4; ++j) dir[j] = 0.0f;
    }
    float* ro = out + (size_t)A + (size_t)a * 8;
#pragma unroll
    for (int j = 0; j < 8; ++j) ro[j] = reg[j];
    float* dro = out + (size_t)A * 9 + (size_t)a * 4;
#pragma unroll
    for (int j = 0; j < 4; ++j) dro[j] = dir[j];
}

// --------------------------- kernel: GT prep -------------------------------
__global__ void __launch_bounds__(MAX_GT)
prep_gt_kernel(const float* __restrict__ gt, int G,
               float* __restrict__ wsgt,
               unsigned long long* __restrict__ winner) {
    int g = threadIdx.x;
    if (g >= G) return;
    float x = gt[g * 7 + 0], y = gt[g * 7 + 1], z = gt[g * 7 + 2];
    float l = gt[g * 7 + 3], w = gt[g * 7 + 4], h = gt[g * 7 + 5];
    float r = gt[g * 7 + 6];
    float c = fabsf(cosf(r)), s = fabsf(sinf(r));
    float ex = 0.5f * (l * c + w * s);
    float ey = 0.5f * (l * s + w * c);
    float x1 = x - ex, y1 = y - ey, x2 = x + ex, y2 = y + ey;
    float area = (x2 - x1 + 1.0f) * (y2 - y1 + 1.0f);
    float* p = wsgt + g * 12;
    p[0] = x1; p[1] = y1; p[2] = x2; p[3] = y2; p[4] = area;
    p[5] = x;  p[6] = y;  p[7] = z;  p[8] = l;  p[9] = w;  p[10] = h; p[11] = r;
    // init winner key: iou = 0.0 (bits 0), idx = 0 encoded as ~0 - 0
    winner[g] = 0xFFFFFFFFull;
}

// --------------------- kernel: force per-GT top-1 anchors ------------------
__global__ void __launch_bounds__(MAX_GT)
finalize_top1_kernel(const float* __restrict__ anchors,
                     const float4* __restrict__ standup,
                     const float* __restrict__ wsgt,
                     int G, int A,
                     const unsigned long long* __restrict__ winner,
                     float* __restrict__ out) {
    int g = blockIdx.x * blockDim.x + threadIdx.x;
    if (g >= G) return;
    unsigned long long key = winner[g];
    unsigned a = 0xFFFFFFFFu - (unsigned)(key & 0xFFFFFFFFull);
    if (a >= (unsigned)A) return;  // safety

    const float4 sb = standup[a];
    const float a1 = (sb.z - sb.x + 1.0f) * (sb.w - sb.y + 1.0f);
    float best = -1.0f;
    int bg = 0;
    for (int gg = 0; gg < G; ++gg) {
        const float* p = wsgt + gg * 12;
        float iou = iou_su(sb.x, sb.y, sb.z, sb.w, a1, p);
        if (iou > best) { best = iou; bg = gg; }
    }
    float reg[8], dir[4];
    encode_one(anchors + (size_t)a * 7, wsgt + bg * 12, reg, dir);

    out[a] = 1.0f;  // pos_mask forced by is_top1
    float* ro = out + (size_t)A + (size_t)a * 8;
#pragma unroll
    for (int j = 0; j < 8; ++j) ro[j] = reg[j];
    float* dro = out + (size_t)A * 9 + (size_t)a * 4;
#pragma unroll
    for (int j = 0; j < 4; ++j) dro[j] = dir[j];
}

// ---------------------------------------------------------------------------
extern "C" void kernel_launch(void* const* d_in, const int* in_sizes, int n_in,
                              void* d_out, int out_size, void* d_ws, size_t ws_size,
                              hipStream_t stream) {
    const float* gt_boxes = (const float*)d_in[0];
    const float* anchors  = (const float*)d_in[1];
    const float* standup  = (const float*)d_in[2];
    float* out = (float*)d_out;

    int G = in_sizes[0] / 7;             // 128
    int A = in_sizes[1] / 7;             // 281600
    if (G > MAX_GT) G = MAX_GT;

    // workspace: [0, 6144) gt table (128*12 f32), [6144, 7168) winner keys (128 u64)
    float* wsgt = (float*)d_ws;
    unsigned long long* winner =
        (unsigned long long*)((char*)d_ws + (size_t)MAX_GT * 12 * sizeof(float));

    prep_gt_kernel<<<1, MAX_GT, 0, stream>>>(gt_boxes, G, wsgt, winner);

    const int threads = 256;  // 8 wave32 waves per workgroup
    const int blocks = (A + threads - 1) / threads;
    assign_main_kernel<<<blocks, threads, 0, stream>>>(
        anchors, (const float4*)standup, wsgt, G, A, winner, out);

    finalize_top1_kernel<<<1, MAX_GT, 0, stream>>>(
        anchors, (const float4*)standup, wsgt, G, A, winner, out);
}